// EKFGradient_45681272160775
// MI455X (gfx1250) — compile-verified
//
#include <hip/hip_runtime.h>

// ---------------------------------------------------------------------------
// EKF scan for MI455X (gfx1250): one wave32, all 6x6 covariance algebra done
// with V_WMMA_F32_16X16X4_F32 (6x6 padded into 16x16 tiles, K padded to 4).
// Sequential 262k-step Riccati/EKF chain -> latency-bound; WMMA collapses each
// small matrix product to one wave instruction; the scalar 3x3 inverse and
// nonlinear predict run on the SALU float pipe (compiler-scalarized) and
// co-execute with the WMMA/DS stream. Loop-invariant operands (F, F^T, Q)
// are hoisted into registers so only P-dependent tiles touch LDS per step.
// ---------------------------------------------------------------------------

typedef float v2f __attribute__((ext_vector_type(2)));
typedef float v8f __attribute__((ext_vector_type(8)));

#define EKF_DT (1.0f / 120.0f)
#define HALF_L (1.948f * 0.5f)
#define HALF_W (1.038f * 0.5f - 0.03165f)
#define TS 17  // LDS tile row stride (floats); 17 avoids 64-bank conflicts

__device__ __forceinline__ v8f wmma_f32(v2f a, v2f b, v8f c) {
  // D = A(16x4) * B(4x16) + C(16x16), f32, wave32
  return __builtin_amdgcn_wmma_f32_16x16x4_f32(false, a, false, b, (short)0, c,
                                               false, false);
}

// A operand (16x4, MxK): lanes 0-15 hold M=lane / K=k0+{0,1}; lanes 16-31 hold
// M=lane-16 / K=k0+{2,3}.
__device__ __forceinline__ v2f load_A(const float* t, int k0, int lane) {
  int m = lane & 15, hi = lane >> 4;
  v2f a;
  a.x = t[m * TS + k0 + 2 * hi];
  a.y = t[m * TS + k0 + 2 * hi + 1];
  return a;
}

// B operand (4x16, KxN): VGPR0 = rows k0+{0|2}, VGPR1 = rows k0+{1|3}, N=lane&15.
__device__ __forceinline__ v2f load_B(const float* t, int k0, int lane) {
  int n = lane & 15, hi = lane >> 4;
  v2f b;
  b.x = t[(k0 + 2 * hi) * TS + n];
  b.y = t[(k0 + 2 * hi + 1) * TS + n];
  return b;
}

// C/D layout (16x16): VGPR r = row (r + 8*hi), col = lane&15.
__device__ __forceinline__ v8f load_C(const float* t, int lane) {
  int n = lane & 15, hi = lane >> 4;
  v8f c;
#pragma unroll
  for (int r = 0; r < 8; ++r) c[r] = t[(r + 8 * hi) * TS + n];
  return c;
}

__device__ __forceinline__ void store_D(float* t, v8f d, int lane) {
  int n = lane & 15, hi = lane >> 4;
#pragma unroll
  for (int r = 0; r < 8; ++r) t[(r + 8 * hi) * TS + n] = d[r];
}

__device__ __forceinline__ float sgn(float x) {
  return (x > 0.f) ? 1.f : ((x < 0.f) ? -1.f : 0.f);
}

__global__ __launch_bounds__(32) void EKFGradient_45681272160775_kernel(
    const float4* __restrict__ dat, const float* __restrict__ params,
    float* __restrict__ out, int T) {
  __shared__ float Ft[16 * TS];   // process Jacobian F (staging only)
  __shared__ float FTt[16 * TS];  // F^T (staging only)
  __shared__ float Qt[16 * TS];   // process noise Q (staging only)
  __shared__ float Pt[16 * TS];   // current covariance P
  __shared__ float Tt[16 * TS];   // temp F*P
  __shared__ float Ppt[16 * TS];  // predicted covariance Pp
  __shared__ float Kt[16 * TS];   // holds -K (6x3)

  const int lane = threadIdx.x;
  const float friction = params[0];
  const float damping = params[1];
  const float restitution = params[2];

  // Zero all tiles (padding regions must be zero for the WMMA padding trick).
  for (int i = lane; i < 16 * TS; i += 32) {
    Ft[i] = 0.f; FTt[i] = 0.f; Qt[i] = 0.f; Pt[i] = 0.f;
    Tt[i] = 0.f; Ppt[i] = 0.f; Kt[i] = 0.f;
  }
  __syncthreads();

  if (lane == 0) {
    const float fd = 1.f - EKF_DT * damping;
    for (int i = 0; i < 6; ++i) Ft[i * TS + i] = 1.f;
    Ft[2 * TS + 2] = fd;
    Ft[3 * TS + 3] = fd;
    Ft[0 * TS + 2] = EKF_DT;
    Ft[1 * TS + 3] = EKF_DT;
    Ft[4 * TS + 5] = EKF_DT;
    for (int i = 0; i < 6; ++i)
      for (int j = 0; j < 6; ++j) FTt[j * TS + i] = Ft[i * TS + j];
    const float qd[6] = {2e-10f, 2e-10f, 3e-7f, 3e-7f, 0.01f, 0.1f};
    for (int i = 0; i < 6; ++i) Qt[i * TS + i] = qd[i];
    for (int i = 0; i < 6; ++i) Pt[i * TS + i] = 0.01f;  // P0
  }
  __syncthreads();

  // Hoist loop-invariant WMMA operands into registers: F as A-operands,
  // F^T as B-operands, Q as the C accumulator seed.
  const v2f F_A0 = load_A(Ft, 0, lane);
  const v2f F_A1 = load_A(Ft, 4, lane);
  const v2f FT_B0 = load_B(FTt, 0, lane);
  const v2f FT_B1 = load_B(FTt, 4, lane);
  const v8f Q_C = load_C(Qt, lane);
  const v8f ZERO_C = {0.f, 0.f, 0.f, 0.f, 0.f, 0.f, 0.f, 0.f};

  // Initial state from first two measurements (all lanes replicated).
  float4 z0 = dat[0], z1 = dat[1];
  z0.x += HALF_L;
  z1.x += HALF_L;
  const float dtm = z1.w - z0.w;
  float s0 = z0.x, s1 = z0.y;
  float s2 = (z1.x - z0.x) / dtm, s3 = (z1.y - z0.y) / dtm;
  float s4 = z0.z, s5 = (z1.z - z0.z) / dtm;
  float acc = 0.f;

  for (int t = 1; t < T; ++t) {
    if (t + 16 < T) __builtin_prefetch(&dat[t + 16], 0, 1);  // global_prefetch_b8
    float4 z = dat[t];
    z.x += HALF_L;

    // ---- nonlinear predict (lane-replicated -> SALU float pipe) ----
    const float dx = s2, dy = s3;
    const float dxn = dx - EKF_DT * (damping * dx + friction * sgn(dx));
    float dyn = dy - EKF_DT * (damping * dy + friction * sgn(dy));
    const float xn = s0 + EKF_DT * dx;
    float yn = s1 + EKF_DT * dy;
    const float thn = s4 + EKF_DT * s5;
    const float wn = s5;
    const bool hit = fabsf(yn) > HALF_W;
    const float syn = sgn(yn);
    yn = hit ? syn * HALF_W : yn;
    dyn = hit ? -restitution * dyn : dyn;

    // ---- Pp = F P F^T + Q : 4 WMMAs (K split 0..3 / 4..7, zero-padded) ----
    v8f fp = ZERO_C;
    fp = wmma_f32(F_A1, load_B(Pt, 4, lane), fp);
    fp = wmma_f32(F_A0, load_B(Pt, 0, lane), fp);
    store_D(Tt, fp, lane);
    __syncthreads();

    v8f pp = Q_C;  // Q as the accumulator seed
    pp = wmma_f32(load_A(Tt, 4, lane), FT_B1, pp);
    pp = wmma_f32(load_A(Tt, 0, lane), FT_B0, pp);
    store_D(Ppt, pp, lane);
    __syncthreads();

    // ---- S = H Pp H^T + R (rows/cols {0,1,4}); 3x3 adjugate inverse ----
    const float s00 = Ppt[0 * TS + 0] + 2.5e-7f;
    const float s01 = Ppt[0 * TS + 1];
    const float s02 = Ppt[0 * TS + 4];
    const float s10 = Ppt[1 * TS + 0];
    const float s11 = Ppt[1 * TS + 1] + 2.5e-7f;
    const float s12 = Ppt[1 * TS + 4];
    const float s20 = Ppt[4 * TS + 0];
    const float s21 = Ppt[4 * TS + 1];
    const float s22 = Ppt[4 * TS + 4] + 0.0091f;

    const float c00 = s11 * s22 - s12 * s21;
    const float c01 = s12 * s20 - s10 * s22;
    const float c02 = s10 * s21 - s11 * s20;
    const float det = s00 * c00 + s01 * c01 + s02 * c02;
    const float inv = 1.f / det;
    const float i00 = c00 * inv, i01 = (s02 * s21 - s01 * s22) * inv,
                i02 = (s01 * s12 - s02 * s11) * inv;
    const float i10 = c01 * inv, i11 = (s00 * s22 - s02 * s20) * inv,
                i12 = (s02 * s10 - s00 * s12) * inv;
    const float i20 = c02 * inv, i21 = (s01 * s20 - s00 * s21) * inv,
                i22 = (s00 * s11 - s01 * s10) * inv;

    // ---- innovation + evaluation term ----
    const float y0 = z.x - xn, y1 = z.y - yn, y2 = z.z - thn;
    const float q0 = i00 * y0 + i01 * y1 + i02 * y2;
    const float q1 = i10 * y0 + i11 * y1 + i12 * y2;
    const float q2 = i20 * y0 + i21 * y1 + i22 * y2;
    acc += det + (y0 * q0 + y1 * q1 + y2 * q2);

    // ---- K = (Pp H^T) Sinv, 18 elements distributed to lanes 0..17; store -K
    if (lane < 18) {
      const int i = lane / 3, j = lane - 3 * (lane / 3);
      const float m0 = Ppt[i * TS + 0];
      const float m1 = Ppt[i * TS + 1];
      const float m2 = Ppt[i * TS + 4];
      const float sj0 = (j == 0) ? i00 : (j == 1) ? i01 : i02;
      const float sj1 = (j == 0) ? i10 : (j == 1) ? i11 : i12;
      const float sj2 = (j == 0) ? i20 : (j == 1) ? i21 : i22;
      Kt[i * TS + j] = -(m0 * sj0 + m1 * sj1 + m2 * sj2);
    }
    __syncthreads();

    // ---- state update: s = pred + K y  (Kt holds -K) ----
    s0 = xn - (Kt[0 * TS + 0] * y0 + Kt[0 * TS + 1] * y1 + Kt[0 * TS + 2] * y2);
    s1 = yn - (Kt[1 * TS + 0] * y0 + Kt[1 * TS + 1] * y1 + Kt[1 * TS + 2] * y2);
    s2 = dxn - (Kt[2 * TS + 0] * y0 + Kt[2 * TS + 1] * y1 + Kt[2 * TS + 2] * y2);
    s3 = dyn - (Kt[3 * TS + 0] * y0 + Kt[3 * TS + 1] * y1 + Kt[3 * TS + 2] * y2);
    s4 = thn - (Kt[4 * TS + 0] * y0 + Kt[4 * TS + 1] * y1 + Kt[4 * TS + 2] * y2);
    s5 = wn - (Kt[5 * TS + 0] * y0 + Kt[5 * TS + 1] * y1 + Kt[5 * TS + 2] * y2);

    // ---- Pn = Pp + (-K) * (H Pp) : one WMMA (K-dim 3, padded to 4) ----
    // B rows: k=0 -> Pp row0, k=1 -> Pp row1, k=2 -> Pp row4, k=3 -> zero.
    v2f bk;
    {
      const int n = lane & 15, hi = lane >> 4;
      const int ka = 2 * hi;      // 0 or 2
      const int kb = 2 * hi + 1;  // 1 or 3
      bk.x = (ka == 2) ? Ppt[4 * TS + n] : Ppt[0 * TS + n];
      bk.y = (kb == 1) ? Ppt[1 * TS + n] : 0.f;
    }
    v8f pn = wmma_f32(load_A(Kt, 0, lane), bk, pp);  // C = Pp (in regs)
    store_D(Pt, pn, lane);
    __syncthreads();
  }

  if (lane == 0) out[0] = acc / (float)(T - 1);
}

extern "C" void kernel_launch(void* const* d_in, const int* in_sizes, int n_in,
                              void* d_out, int out_size, void* d_ws,
                              size_t ws_size, hipStream_t stream) {
  const float4* dat = (const float4*)d_in[0];   // (T, 4) float32 [x,y,th,t]
  const float* params = (const float*)d_in[1];  // [friction, damping, restitution]
  float* out = (float*)d_out;
  const int T = in_sizes[0] / 4;
  hipLaunchKernelGGL(EKFGradient_45681272160775_kernel, dim3(1), dim3(32), 0,
                     stream, dat, params, out, T);
}